// LPGCNGCN_37838661877984
// MI455X (gfx1250) — compile-verified
//
#include <hip/hip_runtime.h>
#include <hip/hip_bf16.h>

typedef float v2f __attribute__((ext_vector_type(2)));
typedef float v8f __attribute__((ext_vector_type(8)));
typedef unsigned int v4u __attribute__((ext_vector_type(4)));
typedef int v8i __attribute__((ext_vector_type(8)));
typedef int v4i __attribute__((ext_vector_type(4)));

#if __has_builtin(__builtin_amdgcn_tensor_load_to_lds) && __has_builtin(__builtin_amdgcn_s_wait_tensorcnt)
#define USE_TDM 1
#else
#define USE_TDM 0
#endif

#if USE_TDM
// 2D f32 tile load via Tensor Data Mover.
//  - rows beyond `rows_avail` zero-fill (hardware OOB handling)
//  - optional LDS padding: pad_amt_code+1... encoding per D# group1:
//      pad_interval code v -> 2^(v+1) DWORDs between pads; pad_amount code v -> v+1 DWORDs
__device__ __forceinline__ void tdm_load_2d(unsigned lds_byte_off, const void* gaddr,
                                            unsigned dim0_elems, unsigned rows_avail,
                                            unsigned tile_rows, unsigned stride_elems,
                                            unsigned pad_int_code, unsigned pad_amt_code,
                                            bool pad_en) {
  unsigned long long ga = (unsigned long long)gaddr;
  v4u g0;
  g0[0] = 1u;                                               // count=1, user descriptor
  g0[1] = lds_byte_off;                                     // D#.lds_addr
  g0[2] = (unsigned)(ga & 0xFFFFFFFFu);                     // global_addr[31:0]
  g0[3] = (unsigned)((ga >> 32) & 0x01FFFFFFu) | 0x80000000u; // addr[56:32] | type=2
  unsigned w0 = (2u << 16);                                 // data_size = 4 bytes
  if (pad_en) w0 |= (1u << 20) | (pad_int_code << 22) | (pad_amt_code << 25);
  v8i g1;
  g1[0] = (int)w0;
  g1[1] = (int)((dim0_elems & 0xFFFFu) << 16);                          // tensor_dim0 lo
  g1[2] = (int)((dim0_elems >> 16) | ((rows_avail & 0xFFFFu) << 16));   // dim0 hi | dim1 lo
  g1[3] = (int)((rows_avail >> 16) | ((dim0_elems & 0xFFFFu) << 16));   // dim1 hi | tile_dim0
  g1[4] = (int)(tile_rows & 0xFFFFu);                                   // tile_dim1 (tile_dim2=0)
  g1[5] = (int)stride_elems;                                            // tensor_dim0_stride lo
  g1[6] = 0;
  g1[7] = 0;
  v4i z4 = {};
#if __clang_major__ >= 23
  v8i z8 = {};
  __builtin_amdgcn_tensor_load_to_lds(g0, g1, z4, z4, z8, 0);
#else
  __builtin_amdgcn_tensor_load_to_lds(g0, g1, z4, z4, 0);
#endif
}
#endif  // USE_TDM

// ======================= degree / GCN norm =======================
__global__ __launch_bounds__(256) void k_init_deg(float* __restrict__ deg, long n) {
  long t = (long)blockIdx.x * 256 + threadIdx.x;
  if (t < n) deg[t] = 1.0f;            // self-loop contribution
}

__global__ __launch_bounds__(256) void k_count_deg(const int* __restrict__ dst,
                                                   float* __restrict__ deg, long e) {
  long t = (long)blockIdx.x * 256 + threadIdx.x;
  if (t < e) atomicAdd(&deg[dst[t]], 1.0f);
}

__global__ __launch_bounds__(256) void k_fin_dinv(float* __restrict__ dinv, long n) {
  long t = (long)blockIdx.x * 256 + threadIdx.x;
  if (t < n) dinv[t] = rsqrtf(dinv[t]);   // deg >= 1 always (self-loop)
}

// ======================= dual GEMM 1: H1 = x@W1, HH1 = x@Wh1 =======================
// x: [N,128] f32, W: [128,64] f32. One block = 16 rows; 8 waves: wave = branch*4 + coltile.
// Each wave: 16x16 output tile, K=128 -> 32x V_WMMA_F32_16X16X4_F32.
// A tile staged by the Tensor Data Mover with HW padding -> LDS stride 132 (conflict-free).
#define ASTR 132   // 128 DW rows + 4 DW pad (TDM pad_interval=6, pad_amount=3)

__global__ __launch_bounds__(256) void k_gemm1_dual(
    const float* __restrict__ x,
    const float* __restrict__ W1, const float* __restrict__ Wh1,
    float* __restrict__ H1, float* __restrict__ HH1, long N) {
  __shared__ float As[16 * ASTR];
  const int tid = threadIdx.x;
  const long row0 = (long)blockIdx.x * 16;

#if USE_TDM
  if ((tid >> 5) == 0) {   // one wave issues the DMA (EXEC ignored by TDM)
    tdm_load_2d((unsigned)(unsigned long long)(uintptr_t)As, x + row0 * 128,
                /*dim0*/128, /*rows_avail*/(unsigned)(N - row0), /*tile_rows*/16,
                /*stride*/128, /*pad_int 2^7=128DW*/6, /*pad 4DW*/3, true);
    __builtin_amdgcn_s_wait_tensorcnt(0);
  }
#else
  for (int idx = tid; idx < 16 * 128; idx += 256) {
    int r = idx >> 7, c = idx & 127;
    long rr = row0 + r; if (rr >= N) rr = N - 1;
    As[r * ASTR + c] = x[rr * 128 + c];
  }
#endif
  __syncthreads();

  const int wave = tid >> 5, lane = tid & 31;
  const float* __restrict__ W = (wave < 4) ? W1 : Wh1;
  float* __restrict__ Hout    = (wave < 4) ? H1 : HH1;
  const int ct = (wave & 3) * 16;     // column tile offset
  const int r  = lane & 15;           // A row / B,C column within tile
  const int hi = lane >> 4;           // half-wave select
  const int kb = hi * 2;              // K sub-offset per ISA 16x4 f32 A layout

  v8f acc = {};
  for (int k0 = 0; k0 < 128; k0 += 4) {
    v2f a, b;
    a.x = As[r * ASTR + k0 + kb];
    a.y = As[r * ASTR + k0 + kb + 1];
    b.x = W[(k0 + kb) * 64 + ct + r];
    b.y = W[(k0 + kb + 1) * 64 + ct + r];
    acc = __builtin_amdgcn_wmma_f32_16x16x4_f32(false, a, false, b,
                                                (short)0, acc, false, false);
  }
#pragma unroll
  for (int v = 0; v < 8; ++v) {
    long row = row0 + v + hi * 8;     // C/D layout: VGPR v -> M = v + 8*hi
    if (row < N) Hout[row * 64 + ct + r] = acc[v];
  }
}

// ======================= layer-1 init: bias + self-loop term =======================
__global__ __launch_bounds__(256) void k_init_l1(
    const float* __restrict__ dinv,
    const float* __restrict__ H1, const float* __restrict__ HH1,
    const float* __restrict__ b1, const float* __restrict__ bh1,
    float* __restrict__ XG1, float* __restrict__ XH1, long NF) {
  long t = (long)blockIdx.x * 256 + threadIdx.x;
  if (t >= NF) return;
  long i = t >> 6; int f = (int)(t & 63);
  float di = dinv[i], w = di * di;    // self-loop norm = dinv[i]^2
  XG1[t] = b1[f] + w * H1[t];
  XH1[t] = bh1[f] + w * HH1[t];
}

// ======================= layer-1 edge scatter (both branches) =======================
// One wave per edge; lane l handles features {2l, 2l+1} (float2-coalesced rows).
__global__ __launch_bounds__(256) void k_scatter_l1(
    const int* __restrict__ src, const int* __restrict__ dst,
    const float* __restrict__ dinv,
    const float* __restrict__ H1, const float* __restrict__ HH1,
    float* __restrict__ XG1, float* __restrict__ XH1, long E) {
  long t = (long)blockIdx.x * 256 + threadIdx.x;
  long e = t >> 5;
  if (e >= E) return;
  int lane = (int)(t & 31);
  int s = src[e], d = dst[e];
  float w = dinv[s] * dinv[d];
  const float2 hg = *(const float2*)(H1  + (long)s * 64 + lane * 2);
  const float2 hh = *(const float2*)(HH1 + (long)s * 64 + lane * 2);
  float* pg = XG1 + (long)d * 64 + lane * 2;
  float* ph = XH1 + (long)d * 64 + lane * 2;
  atomicAdd(pg,     w * hg.x);
  atomicAdd(pg + 1, w * hg.y);
  atomicAdd(ph,     w * hh.x);
  atomicAdd(ph + 1, w * hh.y);
}

// ======================= dual GEMM 2: G2 = relu(XG1)@W2, H2 = relu(XH1)@Wh2 ==========
// [N,64] @ [64,32]. One block = 16 rows; 4 waves: wave = branch*2 + coltile. 16 WMMA/tile.
// Two TDM tile loads (raw); ReLU fused into the fragment read.
#define ASTR2 68   // 64 DW rows + 4 DW pad (TDM pad_interval=5, pad_amount=3)

__global__ __launch_bounds__(128) void k_gemm2_dual(
    const float* __restrict__ XG1, const float* __restrict__ XH1,
    const float* __restrict__ W2, const float* __restrict__ Wh2,
    float* __restrict__ G2, float* __restrict__ H2, long N) {
  __shared__ float As[2 * 16 * ASTR2];
  const int tid = threadIdx.x;
  const long row0 = (long)blockIdx.x * 16;

#if USE_TDM
  if ((tid >> 5) == 0) {
    unsigned base = (unsigned)(unsigned long long)(uintptr_t)As;
    tdm_load_2d(base,                      XG1 + row0 * 64, 64,
                (unsigned)(N - row0), 16, 64, /*2^6=64DW*/5, /*4DW*/3, true);
    tdm_load_2d(base + 16 * ASTR2 * 4,     XH1 + row0 * 64, 64,
                (unsigned)(N - row0), 16, 64, 5, 3, true);
    __builtin_amdgcn_s_wait_tensorcnt(0);
  }
#else
  for (int idx = tid; idx < 2 * 16 * 64; idx += 128) {
    int br = idx >> 10, rem = idx & 1023;
    int r = rem >> 6, c = rem & 63;
    long rr = row0 + r; if (rr >= N) rr = N - 1;
    const float* __restrict__ Xin = br ? XH1 : XG1;
    As[br * 16 * ASTR2 + r * ASTR2 + c] = Xin[rr * 64 + c];
  }
#endif
  __syncthreads();

  const int wave = tid >> 5, lane = tid & 31;
  const int branch = wave >> 1;
  const float* __restrict__ W = branch ? Wh2 : W2;
  float* __restrict__ Out     = branch ? H2 : G2;
  const float* __restrict__ A = As + branch * 16 * ASTR2;
  const int ct = (wave & 1) * 16;
  const int r  = lane & 15;
  const int hi = lane >> 4;
  const int kb = hi * 2;

  v8f acc = {};
  for (int k0 = 0; k0 < 64; k0 += 4) {
    v2f a, b;
    a.x = fmaxf(A[r * ASTR2 + k0 + kb], 0.0f);       // fused ReLU
    a.y = fmaxf(A[r * ASTR2 + k0 + kb + 1], 0.0f);
    b.x = W[(k0 + kb) * 32 + ct + r];
    b.y = W[(k0 + kb + 1) * 32 + ct + r];
    acc = __builtin_amdgcn_wmma_f32_16x16x4_f32(false, a, false, b,
                                                (short)0, acc, false, false);
  }
#pragma unroll
  for (int v = 0; v < 8; ++v) {
    long row = row0 + v + hi * 8;
    if (row < N) Out[row * 32 + ct + r] = acc[v];
  }
}

// ======================= layer-2 init =======================
__global__ __launch_bounds__(256) void k_init_l2(
    const float* __restrict__ dinv,
    const float* __restrict__ G2, const float* __restrict__ H2,
    const float* __restrict__ b2, const float* __restrict__ bh2,
    float* __restrict__ XG2, float* __restrict__ XH2, long NC) {
  long t = (long)blockIdx.x * 256 + threadIdx.x;
  if (t >= NC) return;
  long i = t >> 5; int c = (int)(t & 31);
  float di = dinv[i], w = di * di;
  XG2[t] = b2[c] + w * G2[t];
  XH2[t] = bh2[c] + w * H2[t];
}

// ======================= layer-2 edge scatter =======================
__global__ __launch_bounds__(256) void k_scatter_l2(
    const int* __restrict__ src, const int* __restrict__ dst,
    const float* __restrict__ dinv,
    const float* __restrict__ G2, const float* __restrict__ H2,
    float* __restrict__ XG2, float* __restrict__ XH2, long E) {
  long t = (long)blockIdx.x * 256 + threadIdx.x;
  long e = t >> 5;
  if (e >= E) return;
  int lane = (int)(t & 31);
  int s = src[e], d = dst[e];
  float w = dinv[s] * dinv[d];
  atomicAdd(XG2 + (long)d * 32 + lane, w * G2[(long)s * 32 + lane]);
  atomicAdd(XH2 + (long)d * 32 + lane, w * H2[(long)s * 32 + lane]);
}

// ======================= final: concat GEMV + log-softmax (one warp / node) =========
__global__ __launch_bounds__(256) void k_final_lp(
    const float* __restrict__ XG2, const float* __restrict__ XH2,
    const float* __restrict__ Wlp, const float* __restrict__ blp,
    float* __restrict__ out, long N) {
  __shared__ float Ws[64 * 32];
  const int tid = threadIdx.x;
  for (int idx = tid; idx < 2048; idx += 256) Ws[idx] = Wlp[idx];
  __syncthreads();

  const int warp = tid >> 5, lane = tid & 31;
  long n = (long)blockIdx.x * 8 + warp;
  if (n >= N) return;

  float g = XG2[n * 32 + lane];
  float h = XH2[n * 32 + lane];
  float acc = blp[lane];
#pragma unroll
  for (int k = 0; k < 32; ++k) {
    acc = fmaf(__shfl(g, k, 32), Ws[k * 32 + lane], acc);
    acc = fmaf(__shfl(h, k, 32), Ws[(32 + k) * 32 + lane], acc);
  }
  // numerically stable log-softmax across the 32 lanes (wave32)
  float m = acc;
  for (int off = 16; off; off >>= 1) m = fmaxf(m, __shfl_xor(m, off, 32));
  float ex = expf(acc - m);
  float s = ex;
  for (int off = 16; off; off >>= 1) s += __shfl_xor(s, off, 32);
  out[n * 32 + lane] = (acc - m) - logf(s);
}

// ======================= host launcher =======================
extern "C" void kernel_launch(void* const* d_in, const int* in_sizes, int n_in,
                              void* d_out, int out_size, void* d_ws, size_t ws_size,
                              hipStream_t stream) {
  const float* x   = (const float*)d_in[0];
  const int*   ei  = (const int*)d_in[1];       // [2,E]: row0=src, row1=dst
  // d_in[2] = hyperedge_index: unused by the reference computation
  const float* W1  = (const float*)d_in[3];
  const float* b1  = (const float*)d_in[4];
  const float* Wh1 = (const float*)d_in[5];
  const float* bh1 = (const float*)d_in[6];
  const float* W2  = (const float*)d_in[7];
  const float* b2  = (const float*)d_in[8];
  const float* Wh2 = (const float*)d_in[9];
  const float* bh2 = (const float*)d_in[10];
  const float* Wlp = (const float*)d_in[11];
  const float* blp = (const float*)d_in[12];
  float* out = (float*)d_out;

  const long N = in_sizes[0] / 128;
  const long E = in_sizes[1] / 2;
  const int* esrc = ei;
  const int* edst = ei + E;

  // workspace carving (floats): dinv | H1 | HH1 | XG1 | XH1   (~103 MB)
  float* wsf  = (float*)d_ws;
  float* dinv = wsf;
  float* H1   = dinv + N;
  float* HH1  = H1 + N * 64;
  float* XG1  = HH1 + N * 64;
  float* XH1  = XG1 + N * 64;
  // reuse after their producers are consumed (kernels are stream-ordered):
  float* G2  = H1;   // H1 dead after scatter_l1
  float* H2  = HH1;
  float* XG2 = XG1;  // XG1 dead after gemm2 reads it
  float* XH2 = XH1;

  const unsigned gN   = (unsigned)((N + 255) / 256);
  const unsigned gE   = (unsigned)((E + 255) / 256);
  const unsigned gE32 = (unsigned)((E * 32 + 255) / 256);
  const unsigned gT16 = (unsigned)((N + 15) / 16);

  // 1) GCN normalization
  k_init_deg <<<gN, 256, 0, stream>>>(dinv, N);
  k_count_deg<<<gE, 256, 0, stream>>>(edst, dinv, E);
  k_fin_dinv <<<gN, 256, 0, stream>>>(dinv, N);

  // 2) layer 1: dual GEMM -> init(bias+self-loop) -> edge scatter
  k_gemm1_dual<<<gT16, 256, 0, stream>>>(x, W1, Wh1, H1, HH1, N);
  k_init_l1<<<(unsigned)((N * 64 + 255) / 256), 256, 0, stream>>>(
      dinv, H1, HH1, b1, bh1, XG1, XH1, N * 64);
  k_scatter_l1<<<gE32, 256, 0, stream>>>(esrc, edst, dinv, H1, HH1, XG1, XH1, E);

  // 3) layer 2: dual GEMM (fused ReLU) -> init -> edge scatter
  k_gemm2_dual<<<gT16, 128, 0, stream>>>(XG1, XH1, W2, Wh2, G2, H2, N);
  k_init_l2<<<(unsigned)((N * 32 + 255) / 256), 256, 0, stream>>>(
      dinv, G2, H2, b2, bh2, XG2, XH2, N * 32);
  k_scatter_l2<<<gE32, 256, 0, stream>>>(esrc, edst, dinv, G2, H2, XG2, XH2, E);

  // 4) concat GEMV + log-softmax
  k_final_lp<<<(unsigned)((N + 7) / 8), 256, 0, stream>>>(XG2, XH2, Wlp, blp, out, N);
}